// FeatureHead_6116033429573
// MI455X (gfx1250) — compile-verified
//
#include <hip/hip_runtime.h>
#include <math.h>

typedef __attribute__((ext_vector_type(2))) float v2f;
typedef __attribute__((ext_vector_type(8))) float v8f;
typedef __attribute__((ext_vector_type(4))) unsigned int u32x4;
typedef __attribute__((ext_vector_type(8))) int i32x8;
typedef __attribute__((ext_vector_type(4))) int i32x4;

#define HCN   32
#define CN    64
#define HN    256
#define WN    256
#define BN    8
#define HWN   (HN * WN)

#if defined(__has_builtin)
#  if __has_builtin(__builtin_amdgcn_tensor_load_to_lds) && \
      __has_builtin(__builtin_amdgcn_s_wait_tensorcnt)
#    define USE_TDM 1
#  endif
#endif
#ifndef USE_TDM
#  define USE_TDM 0
#endif

// ---------------------------------------------------------------------------
// Pass 1: fused over color. One block (256 threads, 8 waves) = one (b, row).
//   1) wave 0 launches ONE TDM tensor_load_to_lds: 32 even-channel row
//      segments (256 f32 each, stride 2*H*W elements) -> contiguous LDS.
//   2) all waves compute score via V_WMMA_F32_16X16X4_F32 (global reads only,
//      fully overlapped with the TDM transfer).
//   3) s_wait_tensorcnt 0 + barrier, then 5-tap depthwise conv from LDS,
//      scaled by w_bbx (scale factored out of the conv), max over channels.
//   4) box math; writes score and the UNMASKED [s,x1,y1,x2,y2] tuple.
// ---------------------------------------------------------------------------
__global__ __launch_bounds__(256) void fh_main(const float* __restrict__ color,
                                               const float* __restrict__ w_bbx,
                                               const float* __restrict__ w_width,
                                               const float* __restrict__ w_score,
                                               float* __restrict__ score_out,
                                               float* __restrict__ box_out)
{
    __shared__ float rowbuf[HCN][WN];   // raw even-channel rows (32 KB)

    const int tid = threadIdx.x;
    const int b  = blockIdx.x >> 8;
    const int h  = blockIdx.x & 255;
    const float* __restrict__ crow = color + ((size_t)b * CN) * HWN + (size_t)h * WN;

    // prefetch start of the next row's data stream (global_prefetch_b8)
    __builtin_prefetch(crow + WN, 0, 1);

#if USE_TDM
    if (tid < 32) {   // wave 0 issues the DMA (EXEC ignored by TDM)
        const unsigned long long ga = (unsigned long long)(uintptr_t)crow;
        const unsigned ldsoff = (unsigned)(uintptr_t)(&rowbuf[0][0]); // low 32b = LDS byte offset

        u32x4 g0;
        g0[0] = 1u;                                          // count=1 (valid), user mode
        g0[1] = ldsoff;                                      // lds_addr (bytes)
        g0[2] = (unsigned)(ga & 0xffffffffull);              // global_addr[31:0]
        g0[3] = (unsigned)((ga >> 32) & 0x01ffffffull)       // global_addr[56:32]
              | (2u << 30);                                  // type = 2 ("image")

        i32x8 g1;
        g1[0] = 0x20000;                 // workgroup_mask=0 | data_size=2 (4 bytes)
        g1[1] = (int)(256u << 16);       // atomic_barrier_addr=0 | tensor_dim0[15:0]=256
        g1[2] = (int)(32u << 16);        // tensor_dim0[31:16]=0  | tensor_dim1[15:0]=32
        g1[3] = (int)(256u << 16);       // tensor_dim1[31:16]=0  | tile_dim0=256
        g1[4] = 32;                      // tile_dim1=32 | tile_dim2=0
        g1[5] = 2 * HWN;                 // tensor_dim0_stride[31:0] = 131072 elems
        g1[6] = 0;                       // stride0[47:32]=0 | tensor_dim1_stride[15:0]=0
        g1[7] = 0;

        i32x4 z4 = {0, 0, 0, 0};                     // groups 2/3 unused (2-D tile)
        i32x8 z8 = {0, 0, 0, 0, 0, 0, 0, 0};         // trailing group (clang-23 form)
        __builtin_amdgcn_tensor_load_to_lds(g0, g1, z4, z4, z8, 0);
    }
#else
    #pragma unroll 4
    for (int k = 0; k < HCN; ++k)
        rowbuf[k][tid] = crow[(size_t)(2 * k) * HWN + tid];
#endif

    // ---- score via f32 WMMA (overlaps the TDM transfer) --------------------
    //   A (16x4) = w_score[k..k+3] broadcast over all 16 rows,
    //   B (4x16) = 4 channels x 16 pixels. 16 chained WMMAs cover K=64.
    //   All A rows equal => every D element of a lane equals the score of
    //   pixel (lane&15) of its tile -> extraction independent of row layout.
    const int lane     = tid & 31;
    const int grp      = lane >> 4;   // 0: VGPRs hold K=k0,k0+1 ; 1: K=k0+2,k0+3
    const int col      = lane & 15;   // N (pixel within tile)
    const int waveBase = tid & ~31;   // this wave owns pixels waveBase..waveBase+31

    v8f acc0 = {};   // tile 0: pixels waveBase + 0..15
    v8f acc1 = {};   // tile 1: pixels waveBase + 16..31
    #pragma unroll
    for (int k0 = 0; k0 < CN; k0 += 4) {
        const int kc = k0 + 2 * grp;
        v2f a;  a.x = w_score[kc];  a.y = w_score[kc + 1];
        const float* __restrict__ p0 = crow + (size_t)kc * HWN;
        const float* __restrict__ p1 = crow + (size_t)(kc + 1) * HWN;
        v2f b0;  b0.x = p0[waveBase + col];       b0.y = p1[waveBase + col];
        v2f b1;  b1.x = p0[waveBase + 16 + col];  b1.y = p1[waveBase + 16 + col];
        acc0 = __builtin_amdgcn_wmma_f32_16x16x4_f32(false, a, false, b0,
                                                     (short)0, acc0, false, false);
        acc1 = __builtin_amdgcn_wmma_f32_16x16x4_f32(false, a, false, b1,
                                                     (short)0, acc1, false, false);
    }
    // lane L of grp g holds the score of pixel waveBase + 16*g + col == tid
    const float sc = grp ? acc1[0] : acc0[0];

#if USE_TDM
    __builtin_amdgcn_s_wait_tensorcnt(0);   // TDM data landed in LDS
#endif
    __syncthreads();

    // ---- 5-tap conv + channel max (zero padding via selects) ---------------
    float wmax = -INFINITY;
    #pragma unroll 4
    for (int k = 0; k < HCN; ++k) {
        const float* __restrict__ ww = w_width + k * 5;
        float v0 = (tid >= 2)      ? rowbuf[k][tid - 2] : 0.0f;
        float v1 = (tid >= 1)      ? rowbuf[k][tid - 1] : 0.0f;
        float v2 = rowbuf[k][tid];
        float v3 = (tid <= WN - 2) ? rowbuf[k][tid + 1] : 0.0f;
        float v4 = (tid <= WN - 3) ? rowbuf[k][tid + 2] : 0.0f;
        float acc = v0 * ww[0];
        acc = fmaf(v1, ww[1], acc);
        acc = fmaf(v2, ww[2], acc);
        acc = fmaf(v3, ww[3], acc);
        acc = fmaf(v4, ww[4], acc);
        acc *= w_bbx[2 * k];               // scale factored out of the conv
        wmax = fmaxf(wmax, acc);
    }

    // ---- box math (mask applied in pass 3) ---------------------------------
    const float s   = 1.0f / (1.0f + __expf(-sc));
    const float cw  = __expf(sc)   * 10.0f;   // wv is score channel
    const float chh = __expf(wmax) * 10.0f;   // hv is wmap channel
    const float xg = (float)tid;
    const float yg = (float)h;

    float x1 = floorf(xg - cw), x2 = ceilf(xg + cw);
    if ((x1 < 0.0f) || (x2 > (float)WN)) {
        float hw = fminf((float)WN - xg, xg);
        x1 = floorf(xg - hw);  x2 = ceilf(xg + hw);
    }
    float y1 = floorf(yg - chh), y2 = ceilf(yg + chh);
    if ((y1 < 0.0f) || (y2 > (float)HN)) {
        float hh = fminf((float)HN - yg, yg);
        y1 = floorf(yg - hh);  y2 = ceilf(yg + hh);
    }

    const size_t p = (size_t)h * WN + tid;
    score_out[(size_t)b * HWN + p] = sc;
    float* __restrict__ ob = box_out + ((size_t)b * HWN + p) * 5;
    ob[0] = s;  ob[1] = x1;  ob[2] = y1;  ob[3] = x2;  ob[4] = y2;
}

// ---------------------------------------------------------------------------
// Pass 2: per-batch 10th-largest sigmoid score (top-k membership threshold).
// ---------------------------------------------------------------------------
__global__ __launch_bounds__(256) void fh_topk(const float* __restrict__ box_out,
                                               float* __restrict__ thresh)
{
    const int b = blockIdx.x, tid = threadIdx.x;
    const float* __restrict__ sb = box_out + (size_t)b * HWN * 5;

    float t[10];
    #pragma unroll
    for (int j = 0; j < 10; ++j) t[j] = -INFINITY;

    for (int i = tid; i < HWN; i += 256) {
        float v = sb[(size_t)i * 5];
        if (v > t[9]) {
            t[9] = v;
            #pragma unroll
            for (int j = 9; j > 0; --j) {     // one bubble pass restores order
                float hi = fmaxf(t[j - 1], t[j]);
                float lo = fminf(t[j - 1], t[j]);
                t[j - 1] = hi;  t[j] = lo;
            }
        }
    }

    __shared__ float cand[256 * 10];
    #pragma unroll
    for (int j = 0; j < 10; ++j) cand[tid * 10 + j] = t[j];
    __syncthreads();

    if (tid == 0) {
        float u[10];
        #pragma unroll
        for (int j = 0; j < 10; ++j) u[j] = -INFINITY;
        for (int i = 0; i < 256 * 10; ++i) {
            float v = cand[i];
            if (v > u[9]) {
                u[9] = v;
                #pragma unroll
                for (int j = 9; j > 0; --j) {
                    float hi = fmaxf(u[j - 1], u[j]);
                    float lo = fminf(u[j - 1], u[j]);
                    u[j - 1] = hi;  u[j] = lo;
                }
            }
        }
        thresh[b] = u[9];   // 10th largest
    }
}

// ---------------------------------------------------------------------------
// Pass 3: m = (s >= t10 & s > 0.6) | (s > 0.8); mask the tuple, emit m.
// ---------------------------------------------------------------------------
__global__ __launch_bounds__(256) void fh_finalize(float* __restrict__ box_out,
                                                   float* __restrict__ m_out,
                                                   const float* __restrict__ thresh)
{
    const int g = blockIdx.x * 256 + threadIdx.x;    // 0 .. B*HW-1
    const int b = g >> 16;
    const float thr = thresh[b];
    float* __restrict__ ob = box_out + (size_t)g * 5;
    const float s = ob[0];
    const bool m = ((s >= thr) && (s > 0.6f)) || (s > 0.8f);
    const float mf = m ? 1.0f : 0.0f;
    ob[0] = s * mf;
    ob[1] *= mf;  ob[2] *= mf;  ob[3] *= mf;  ob[4] *= mf;
    m_out[g] = mf;
}

// ---------------------------------------------------------------------------
extern "C" void kernel_launch(void* const* d_in, const int* in_sizes, int n_in,
                              void* d_out, int out_size, void* d_ws, size_t ws_size,
                              hipStream_t stream) {
    // setup_inputs order: mask(unused), color, w_bbx, w_width, w_height(dead), w_score
    const float* color   = (const float*)d_in[1];
    const float* w_bbx   = (const float*)d_in[2];
    const float* w_width = (const float*)d_in[3];
    const float* w_score = (const float*)d_in[5];

    float* out       = (float*)d_out;
    float* score_out = out;                               // B*H*W
    float* box_out   = out + (size_t)BN * HWN;            // B*H*W*5
    float* m_out     = box_out + (size_t)BN * HWN * 5;    // B*H*W
    float* thresh    = (float*)d_ws;                      // 8 floats

    fh_main<<<BN * HN, 256, 0, stream>>>(color, w_bbx, w_width, w_score,
                                         score_out, box_out);
    fh_topk<<<BN, 256, 0, stream>>>(box_out, thresh);
    fh_finalize<<<(BN * HWN) / 256, 256, 0, stream>>>(box_out, m_out, thresh);
}